// ComSimModel_3169685864864
// MI455X (gfx1250) — compile-verified
//
#include <hip/hip_runtime.h>
#include <hip/hip_bf16.h>

// Problem constants (match reference)
#define TT 4096   // tgt_len
#define BB 64     // bsz
#define DD 256    // embed dim
#define TSTRIDE 260  // padded LDS row stride in floats (bank-conflict-free pattern)

typedef __attribute__((ext_vector_type(2))) float v2f;
typedef __attribute__((ext_vector_type(8))) float v8f;
typedef __attribute__((__vector_size__(4 * sizeof(int)))) int vi4;

#if __has_builtin(__builtin_amdgcn_global_load_async_to_lds_b128)
#define HAVE_ASYNC_LDS 1
#endif

static __device__ __forceinline__ void wait_async0() {
#if __has_builtin(__builtin_amdgcn_s_wait_asynccnt)
    __builtin_amdgcn_s_wait_asynccnt(0);
#else
    asm volatile("s_wait_asynccnt 0" ::: "memory");
#endif
}

// ---------------------------------------------------------------------------
// Kernel 1: per-batch last-occurrence search, gather, 256x256 matvec.
//   x[b,e] = ( (has_one ? graph_dict[node,:] : 0) . W0[e,:] + b0[e] ) * (1/16)
// Also writes gate_x.
// ---------------------------------------------------------------------------
__global__ __launch_bounds__(256) void prep_kernel(
    const long long* __restrict__ batch_nodes,  // [B,T] int64
    const int*       __restrict__ exe,          // [B,T] int32
    const float*     __restrict__ graph_dict,   // [N,D]
    const float*     __restrict__ W0,           // [D,D] (row e, col d)
    const float*     __restrict__ b0,           // [D]
    float*           __restrict__ xout,         // [B,D] workspace
    float*           __restrict__ gate_out)     // [B]
{
    __shared__ int   red[256];
    __shared__ float gvec[DD];
    __shared__ int   s_node;
    __shared__ int   s_has;

    const int b   = blockIdx.x;
    const int tid = threadIdx.x;

    // Last position where exe == 1 (max of masked positions)
    int best = -1;
    #pragma unroll
    for (int i = 0; i < TT / 256; ++i) {
        int t = tid * (TT / 256) + i;
        int e = exe[(size_t)b * TT + t];
        if (e == 1 && t > best) best = t;
    }
    red[tid] = best;
    __syncthreads();
    for (int s = 128; s > 0; s >>= 1) {
        if (tid < s) red[tid] = max(red[tid], red[tid + s]);
        __syncthreads();
    }
    if (tid == 0) {
        int doc = red[0];
        s_has   = (doc >= 0) ? 1 : 0;
        int sd  = (doc < 0) ? 0 : doc;
        s_node  = (int)batch_nodes[(size_t)b * TT + sd];
        gate_out[b] = s_has ? 1.0f : 0.0f;
    }
    __syncthreads();

    // Stage gathered (or zero) embedding row
    gvec[tid] = s_has ? graph_dict[(size_t)s_node * DD + tid] : 0.0f;
    __syncthreads();

    // x[e] = (gvec . W0[e,:] + b0[e]) * scale, scale = 1/sqrt(256) = 1/16
    float acc = 0.0f;
    const float* wrow = W0 + (size_t)tid * DD;
    #pragma unroll 8
    for (int d = 0; d < DD; ++d) acc = fmaf(gvec[d], wrow[d], acc);
    xout[b * DD + tid] = (acc + b0[tid]) * 0.0625f;
}

// ---------------------------------------------------------------------------
// Kernel 2: out[b,t] = x[b,:] . src[t,b,:]   (src streamed once, HBM-bound)
// Each wave: four consecutive (b, 16-t) tiles. Stage 16x256 f32 rows straight
// into LDS via GLOBAL_LOAD_ASYNC_TO_LDS_B128 (padded stride 260 floats ->
// conflict-free ds_load_b64 pattern). Then 64 x V_WMMA_F32_16X16X4_F32 with
//   A = x[b] broadcast over M (16x4),  B = src tile (4x16, N = t).
// All C rows identical -> lanes 0..15 store c[0].
// ---------------------------------------------------------------------------
__global__ __launch_bounds__(64) void sim_kernel(
    const float* __restrict__ src,   // [T,B,D]
    const float* __restrict__ xmat,  // [B,D]
    float*       __restrict__ out)   // [B,T]
{
    __shared__ float sh[2 * 16 * TSTRIDE];   // 2 waves/block, 16 rows each

    const int lane  = threadIdx.x & 31;
    const int wv    = threadIdx.x >> 5;
    const int gwave = blockIdx.x * 2 + wv;   // 0..4095
    float* tile = &sh[wv * 16 * TSTRIDE];

    const int koff = (lane >> 4) << 1;       // lanes 0-15: K+0,K+1 ; 16-31: K+2,K+3
    const int nrow = lane & 15;

    for (int it = 0; it < 4; ++it) {
        const int tl = gwave * 4 + it;       // tile id, 0..16383 (same b across it)
        const int b  = tl >> 8;              // batch
        const int t0 = (tl & 255) << 4;      // first t of tile

        // ---- stage 16 rows x 256 floats into LDS ----
        #pragma unroll
        for (int j = 0; j < 32; ++j) {
            int flat = (j * 32 + lane) * 4;  // 0..4095 floats
            int row  = flat >> 8;
            int col  = flat & 255;
            const float* gp = src + ((size_t)(t0 + row) * BB + b) * DD + col;
            float*       lp = tile + row * TSTRIDE + col;
#ifdef HAVE_ASYNC_LDS
            __builtin_amdgcn_global_load_async_to_lds_b128(
                (vi4*)gp, (vi4*)lp, 0, 0);
#else
            *(float4*)lp = *(const float4*)gp;
#endif
        }
#ifdef HAVE_ASYNC_LDS
        wait_async0();
#endif
        __syncthreads();

        // ---- K-loop: 64 WMMA f32 16x16x4 ----
        const float* xb = xmat + b * DD;
        v8f c = {0.f, 0.f, 0.f, 0.f, 0.f, 0.f, 0.f, 0.f};
        #pragma unroll
        for (int k0 = 0; k0 < DD; k0 += 4) {
            v2f a  = *(const v2f*)(xb + k0 + koff);                      // A: x broadcast
            v2f bm = *(const v2f*)(tile + nrow * TSTRIDE + k0 + koff);   // B: src tile
            c = __builtin_amdgcn_wmma_f32_16x16x4_f32(
                    false, a, false, bm, (short)0, c, false, false);
        }
        if (lane < 16) out[(size_t)b * TT + t0 + lane] = c[0];
        __syncthreads();
    }
}

// ---------------------------------------------------------------------------
extern "C" void kernel_launch(void* const* d_in, const int* in_sizes, int n_in,
                              void* d_out, int out_size, void* d_ws, size_t ws_size,
                              hipStream_t stream) {
    const long long* batch_nodes = (const long long*)d_in[0];
    const int*       exe         = (const int*)d_in[1];
    const float*     src         = (const float*)d_in[2];
    const float*     graph_dict  = (const float*)d_in[3];
    const float*     W0          = (const float*)d_in[4];
    const float*     b0          = (const float*)d_in[5];

    float* out  = (float*)d_out;        // [B*T cosine][B gate]
    float* xws  = (float*)d_ws;         // [B,D] scaled x vectors

    prep_kernel<<<BB, 256, 0, stream>>>(batch_nodes, exe, graph_dict, W0, b0,
                                        xws, out + (size_t)BB * TT);
    sim_kernel<<<2048, 64, 0, stream>>>(src, xws, out);
}